// AlignmentContrastiveLoss_13700945674951
// MI455X (gfx1250) — compile-verified
//
#include <hip/hip_runtime.h>
#include <math.h>

typedef __attribute__((ext_vector_type(2))) float v2f;
typedef __attribute__((ext_vector_type(8))) float v8f;

#define LDA   68      // LDS row stride in floats: 68*4B -> 4-bank rotation per row, conflict-free b64 reads
#define KC    64      // K-chunk staged in LDS
#define MARGIN 0.2f

// im_set: (128, 37, 1024) f32   -> im = im_set[:,1:,:]  (36 rows used, padded to 48)
// s_seq : (128, 53, 1024) f32   -> s  = s_seq[:,1:-2,:] (50 rows used, padded to 64)
// scores: (128,128) f32 in d_ws
__global__ __launch_bounds__(128)
void align_score_kernel(const float* __restrict__ im_set, const float* __restrict__ s_seq,
                        const int* __restrict__ im_len, const int* __restrict__ s_len,
                        float* __restrict__ scores)
{
    __shared__ float ldsA[48 * LDA];
    __shared__ float ldsB[64 * LDA];
    __shared__ float wsum[4];

    const int b    = blockIdx.x;
    const int c    = blockIdx.y;
    const int tid  = threadIdx.x;
    const int lane = tid & 31;
    const int wave = tid >> 5;

    const float* imb = im_set + ((size_t)b * 37 + 1) * 1024;  // row i -> imb + i*1024 (i clamped <= 35)
    const float* scp = s_seq  + ((size_t)c * 53 + 1) * 1024;  // row j -> scp + j*1024 (j clamped <= 49)

    v8f acc0 = {}; v8f acc1 = {}; v8f acc2 = {};

    const int jt   = wave;            // this wave's N tile (j = jt*16 + lane&15)
    const int ln15 = lane & 15;
    const int koff = (lane >> 4) * 2; // A/B 32-bit fragment layout: lanes 16-31 hold K+2,K+3

    for (int k0 = 0; k0 < 1024; k0 += KC) {
        // stage A panel: 48 rows x 64 cols (float4 per thread x6)
        #pragma unroll
        for (int r = 0; r < 6; ++r) {
            int idx  = tid + r * 128;        // float4 index 0..767
            int row  = idx >> 4;             // 16 float4 per row
            int c4   = (idx & 15) << 2;
            int srow = row < 36 ? row : 35;  // clamp: padded rows masked later
            float4 v = *(const float4*)(imb + (size_t)srow * 1024 + k0 + c4);
            *(float4*)&ldsA[row * LDA + c4] = v;
        }
        // stage B panel: 64 rows x 64 cols (float4 per thread x8)
        #pragma unroll
        for (int r = 0; r < 8; ++r) {
            int idx  = tid + r * 128;        // float4 index 0..1023
            int row  = idx >> 4;
            int c4   = (idx & 15) << 2;
            int srow = row < 50 ? row : 49;
            float4 v = *(const float4*)(scp + (size_t)srow * 1024 + k0 + c4);
            *(float4*)&ldsB[row * LDA + c4] = v;
        }
        __syncthreads();

        const float* Bb = &ldsB[(jt * 16 + ln15) * LDA + koff];
        const float* A0 = &ldsA[( 0 + ln15) * LDA + koff];
        const float* A1 = &ldsA[(16 + ln15) * LDA + koff];
        const float* A2 = &ldsA[(32 + ln15) * LDA + koff];
        #pragma unroll
        for (int kk = 0; kk < KC; kk += 4) {
            v2f bf = *(const v2f*)(Bb + kk);
            v2f a0 = *(const v2f*)(A0 + kk);
            v2f a1 = *(const v2f*)(A1 + kk);
            v2f a2 = *(const v2f*)(A2 + kk);
            acc0 = __builtin_amdgcn_wmma_f32_16x16x4_f32(false, a0, false, bf, (short)0, acc0, false, false);
            acc1 = __builtin_amdgcn_wmma_f32_16x16x4_f32(false, a1, false, bf, (short)0, acc1, false, false);
            acc2 = __builtin_amdgcn_wmma_f32_16x16x4_f32(false, a2, false, bf, (short)0, acc2, false, false);
        }
        __syncthreads();
    }

    // Epilogue: replicate reference masking exactly, then max over i, sum over j.
    const int im_l = im_len[b] - 1;   // valid i count (4..36)
    const int s_l  = s_len[c] - 3;    // valid j count (5..50)
    const int j    = jt * 16 + ln15;
    const int ib   = (lane >> 4) * 8; // C/D layout: lanes>=16 hold M = vgpr+8

    float colmax = -INFINITY;
    #pragma unroll
    for (int v = 0; v < 8; ++v) {
        int i0 =  0 + ib + v;
        int i1 = 16 + ib + v;
        int i2 = 32 + ib + v;
        float x0 = acc0[v], x1 = acc1[v], x2 = acc2[v];
        // pad rows (i>=36): exclude from max entirely (-inf); masked valid-range rows: 0 (as reference)
        x0 = (i0 >= im_l || j >= s_l) ? 0.0f : x0;                       // i0 < 36 always
        x1 = (i1 >= im_l || j >= s_l) ? 0.0f : x1;                       // i1 < 36 always
        x2 = (i2 >= 36) ? -INFINITY : ((i2 >= im_l || j >= s_l) ? 0.0f : x2);
        colmax = fmaxf(colmax, fmaxf(x0, fmaxf(x1, x2)));
    }
    // merge the two M-halves that live in lanes L and L^16 (same j)
    colmax = fmaxf(colmax, __shfl_xor(colmax, 16, 32));

    float contrib = (lane < 16 && j < 50) ? colmax : 0.0f;
    #pragma unroll
    for (int off = 16; off >= 1; off >>= 1)
        contrib += __shfl_xor(contrib, off, 32);

    if (lane == 0) wsum[wave] = contrib;
    __syncthreads();
    if (tid == 0)
        scores[(size_t)b * 128 + c] = wsum[0] + wsum[1] + wsum[2] + wsum[3];
}

// hinge-loss reduction: scalar out
__global__ __launch_bounds__(256)
void hinge_loss_kernel(const float* __restrict__ scores, float* __restrict__ out)
{
    __shared__ float diag[128];
    __shared__ float red[256];
    const int tid = threadIdx.x;
    if (tid < 128) diag[tid] = scores[tid * 129];
    __syncthreads();

    float sum = 0.f;
    for (int idx = tid; idx < 128 * 128; idx += 256) {
        int bb = idx >> 7, cc = idx & 127;
        if (bb != cc) {
            float v = scores[idx];
            sum += fmaxf(0.f, MARGIN + v - diag[bb])   // cost_s
                 + fmaxf(0.f, MARGIN + v - diag[cc]);  // cost_im
        }
    }
    red[tid] = sum;
    __syncthreads();
    #pragma unroll
    for (int s = 128; s >= 1; s >>= 1) {
        if (tid < s) red[tid] += red[tid + s];
        __syncthreads();
    }
    if (tid == 0) out[0] = red[0];
}

extern "C" void kernel_launch(void* const* d_in, const int* in_sizes, int n_in,
                              void* d_out, int out_size, void* d_ws, size_t ws_size,
                              hipStream_t stream) {
    const float* im_set = (const float*)d_in[0];
    const float* s_seq  = (const float*)d_in[1];
    const int*   im_len = (const int*)d_in[2];
    const int*   s_len  = (const int*)d_in[3];
    float* scores = (float*)d_ws;   // 128*128 floats

    dim3 grid(128, 128);
    align_score_kernel<<<grid, 128, 0, stream>>>(im_set, s_seq, im_len, s_len, scores);
    hinge_loss_kernel<<<1, 256, 0, stream>>>(scores, (float*)d_out);
}